// SelfAttentionMultiHead_78907139162215
// MI455X (gfx1250) — compile-verified
//
#include <hip/hip_runtime.h>

// CDNA5 / gfx1250, wave32. Fully fused instance-norm + QKV + linear-attn + proj
// using V_WMMA_F32_16X16X4_F32 (full fp32 matrix pipe).

typedef __attribute__((ext_vector_type(2))) float v2f;
typedef __attribute__((ext_vector_type(8))) float v8f;

#define LDST 66   // padded LDS row stride (floats): conflict-free, keeps 8B align

__global__ __launch_bounds__(128)
void fused_linattn_kernel(const float* __restrict__ inp,
                          const float* __restrict__ qkv_w,   // (192,64) row-major
                          const float* __restrict__ qkv_b,   // (192,)
                          const float* __restrict__ proj_w,  // (64,64) row-major
                          const float* __restrict__ proj_b,  // (64,)
                          float* __restrict__ out)           // (25,256,8,8) == input layout
{
    __shared__ float xs [16 * LDST];   // normalized input tile (16 x 64)
    __shared__ float qs [16 * LDST];   // q tile
    __shared__ float ks_[16 * LDST];   // k tile
    __shared__ float vs [16 * LDST];   // v tile
    __shared__ float ats[16 * LDST];   // attn tile (proj input)
    __shared__ float psum[128], psq[128];
    __shared__ float smean[8], srstd[8];
    __shared__ float a2s[32];          // per-instance 2x2 q.k matrices

    const int tid = threadIdx.x;
    const int nb  = blockIdx.x >> 4;        // batch index 0..24
    const int s0  = (blockIdx.x & 15) << 3; // s-block base 0..120 (8 instances)

    // ---------------- load + instance-norm ----------------
    // row m = local_instance*2 + channel ; thread t handles 8 contiguous cols
    const int m    = tid >> 3;        // 0..15
    const int seg  = tid & 7;         // 8-float segment
    const int inst = m >> 1, ch = m & 1;
    const int gbase = ((nb * 2 + ch) * 128 + (s0 + inst)) * 64 + seg * 8;

    float x[8];
    {
        const float4 r0 = *(const float4*)(inp + gbase);
        const float4 r1 = *(const float4*)(inp + gbase + 4);
        x[0] = r0.x; x[1] = r0.y; x[2] = r0.z; x[3] = r0.w;
        x[4] = r1.x; x[5] = r1.y; x[6] = r1.z; x[7] = r1.w;
    }
    float s = 0.f, sq = 0.f;
#pragma unroll
    for (int i = 0; i < 8; i++) { s += x[i]; sq += x[i] * x[i]; }
    psum[tid] = s; psq[tid] = sq;
    __syncthreads();

    if (tid < 8) {  // one thread per instance: reduce 16 partials (= 128 elems)
        float ts = 0.f, tq = 0.f;
#pragma unroll
        for (int p = 0; p < 16; p++) { ts += psum[tid * 16 + p]; tq += psq[tid * 16 + p]; }
        const float mean = ts * (1.f / 128.f);
        const float var  = tq * (1.f / 128.f) - mean * mean;  // biased var
        smean[tid] = mean;
        srstd[tid] = rsqrtf(var + 1e-5f);
    }
    __syncthreads();
    {
        const float mean = smean[inst], rstd = srstd[inst];
#pragma unroll
        for (int i = 0; i < 8; i++)
            xs[m * LDST + seg * 8 + i] = (x[i] - mean) * rstd;
    }
    __syncthreads();

    // ---------------- QKV projection: (16x64) @ (64x192) via WMMA f32 ----------------
    const int wv    = tid >> 5;       // wave 0..3
    const int lane  = tid & 31;
    const int fcol  = lane & 15;      // N column (B/C/D) == M row (A)
    const int khalf = lane >> 4;      // K half select
    const int koff  = khalf * 2;

    const int nq = wv * 16 + fcol;    // column within each of q/k/v (and proj out)

    v8f cq, ck, cv;
    {
        const float bq = qkv_b[nq];
        const float bk = qkv_b[64 + nq];
        const float bv = qkv_b[128 + nq];
#pragma unroll
        for (int r = 0; r < 8; r++) { cq[r] = bq; ck[r] = bk; cv[r] = bv; }
    }
    const float* wqr = qkv_w + (size_t)nq * 64;          // q weight row
    const float* wkr = qkv_w + (size_t)(64 + nq) * 64;   // k weight row
    const float* wvr = qkv_w + (size_t)(128 + nq) * 64;  // v weight row

#pragma unroll
    for (int kb = 0; kb < 64; kb += 4) {
        const v2f a  = *(const v2f*)(&xs[fcol * LDST + kb + koff]); // A: 16x4 fp32
        const v2f b0 = *(const v2f*)(wqr + kb + koff);              // B: 4x16 fp32
        const v2f b1 = *(const v2f*)(wkr + kb + koff);
        const v2f b2 = *(const v2f*)(wvr + kb + koff);
        cq = __builtin_amdgcn_wmma_f32_16x16x4_f32(false, a, false, b0, (short)0, cq, false, false);
        ck = __builtin_amdgcn_wmma_f32_16x16x4_f32(false, a, false, b1, (short)0, ck, false, false);
        cv = __builtin_amdgcn_wmma_f32_16x16x4_f32(false, a, false, b2, (short)0, cv, false, false);
    }

    // spill q/k/v fragments to LDS (D layout: vgpr r -> row r (+8 hi-lanes), lane&15 -> col)
    {
        const int rowoff = khalf * 8;
#pragma unroll
        for (int r = 0; r < 8; r++) {
            qs [(r + rowoff) * LDST + nq] = cq[r];
            ks_[(r + rowoff) * LDST + nq] = ck[r];
            vs [(r + rowoff) * LDST + nq] = cv[r];
        }
    }
    __syncthreads();

    // ---------------- 2x2 attention matrices: A2[i][a][b] = q_row(2i+a) . k_row(2i+b) ----------------
    // Exploits c==2: q @ (k^T @ v) == (q k^T) @ v, a rank-2 combine.
    if (tid < 32) {
        const int i = tid >> 2, aa = (tid >> 1) & 1, bb = tid & 1;
        const float* qr = &qs [(2 * i + aa) * LDST];
        const float* kr = &ks_[(2 * i + bb) * LDST];
        float d = 0.f;
#pragma unroll
        for (int j = 0; j < 64; j++) d += qr[j] * kr[j];
        a2s[tid] = d;
    }
    __syncthreads();

    // ---------------- attn = v + A2 @ v ----------------
    {
        const float w0 = a2s[inst * 4 + ch * 2 + 0];
        const float w1 = a2s[inst * 4 + ch * 2 + 1];
        const float* v0 = &vs[(2 * inst)     * LDST];
        const float* v1 = &vs[(2 * inst + 1) * LDST];
#pragma unroll
        for (int i = 0; i < 8; i++) {
            const int j = seg * 8 + i;
            ats[m * LDST + j] = vs[m * LDST + j] + w0 * v0[j] + w1 * v1[j];
        }
    }
    __syncthreads();

    // ---------------- output projection: (16x64) @ (64x64) via WMMA f32 ----------------
    v8f cp;
    {
        const float bp = proj_b[nq];
#pragma unroll
        for (int r = 0; r < 8; r++) cp[r] = bp;
    }
    const float* wpr = proj_w + (size_t)nq * 64;
#pragma unroll
    for (int kb = 0; kb < 64; kb += 4) {
        const v2f a = *(const v2f*)(&ats[fcol * LDST + kb + koff]);
        const v2f b = *(const v2f*)(wpr + kb + koff);
        cp = __builtin_amdgcn_wmma_f32_16x16x4_f32(false, a, false, b, (short)0, cp, false, false);
    }

    // ---------------- residual add + store (output layout == input layout) ----------------
    {
        const int rowoff = khalf * 8;
#pragma unroll
        for (int r = 0; r < 8; r++) {
            const int mm = r + rowoff;
            const int oi = mm >> 1, oc = mm & 1;
            const size_t gi = ((size_t)(nb * 2 + oc) * 128 + (s0 + oi)) * 64 + nq;
            out[gi] = cp[r] + inp[gi];
        }
    }
}

extern "C" void kernel_launch(void* const* d_in, const int* in_sizes, int n_in,
                              void* d_out, int out_size, void* d_ws, size_t ws_size,
                              hipStream_t stream) {
    const float* inp    = (const float*)d_in[0];
    const float* qkv_w  = (const float*)d_in[1];
    const float* qkv_b  = (const float*)d_in[2];
    const float* proj_w = (const float*)d_in[3];
    const float* proj_b = (const float*)d_in[4];
    float* out = (float*)d_out;

    // 25 batches * 16 s-blocks (8 instances each) = 400 workgroups, 4 waves each
    fused_linattn_kernel<<<dim3(400), dim3(128), 0, stream>>>(
        inp, qkv_w, qkv_b, proj_w, proj_b, out);
}